// Cross_Attention_Local_5952824672975
// MI455X (gfx1250) — compile-verified
//
#include <hip/hip_runtime.h>

typedef __attribute__((ext_vector_type(2))) float v2f;
typedef __attribute__((ext_vector_type(8))) float v8f;

#define NBP     3456      // B * GRID = 128 * 27
#define CDIM    64
#define HEADS   8
#define CH      512       // CDIM * HEADS
#define EPSV    1e-5f

// LDS strides (in floats)
#define EMB_RS  65        // emb tile row stride
#define QKV_RS  72        // per-head q/k/v row stride (16 rows)
#define QKV_HS  1160      // 16*72 + 8  (head stride; +8 breaks bank aliasing of head index)
#define X_RS    524       // X (ctx, (c heads) interleaved) row stride

// pool offsets (floats)
#define OFF_EMB   0
#define OFF_Q     1040                    // 16*65
#define OFF_K     (OFF_Q + 8*QKV_HS)     // + 9280
#define OFF_V     (OFF_K + 8*QKV_HS)
#define OFF_X     (OFF_V + 8*QKV_HS)     // 28880
#define POOL_SZ   (OFF_X + 16*X_RS)      // 37264 floats ~= 145.6 KB
#define OFF_PART  0                       // 8*16*64 = 8192 floats, aliases emb+q (dead by phase 3)

static __device__ __forceinline__ v8f wmma4(v2f a, v2f b, v8f c) {
    return __builtin_amdgcn_wmma_f32_16x16x4_f32(false, a, false, b, (short)0, c, false, false);
}

__global__ __launch_bounds__(256)
void ca_fused(const float* __restrict__ emb, const float* __restrict__ Wq,
              const float* __restrict__ Wk,  const float* __restrict__ Wv,
              const float* __restrict__ Wo,  float* __restrict__ out)
{
    __shared__ __align__(16) float s_pool[POOL_SZ];
    float* s_emb = s_pool + OFF_EMB;
    float* s_q   = s_pool + OFF_Q;
    float* s_k   = s_pool + OFF_K;
    float* s_v   = s_pool + OFF_V;
    float* s_x   = s_pool + OFF_X;

    const int tid  = threadIdx.x;
    const int w    = tid >> 5;     // wave id == head
    const int lane = tid & 31;
    const int lm   = lane & 15;
    const int hf   = lane >> 4;    // lane half

    // -------- phase 0: stage emb tile (2 bp = 16 tokens x 64) --------
    {
        const float* src = emb + (size_t)blockIdx.x * (2 * 8 * CDIM);
        int row = tid >> 4;                // 0..15
        int c0  = (tid & 15) * 4;
        float4 t = *(const float4*)(src + row * CDIM + c0);
        s_emb[row * EMB_RS + c0 + 0] = t.x;
        s_emb[row * EMB_RS + c0 + 1] = t.y;
        s_emb[row * EMB_RS + c0 + 2] = t.z;
        s_emb[row * EMB_RS + c0 + 3] = t.w;
    }
    __syncthreads();

    const v8f z8 = {0.f,0.f,0.f,0.f,0.f,0.f,0.f,0.f};

    // -------- phase 1: q,k,v projections (16x512) = emb(16x64) @ W(64x512) --------
    // wave w owns columns j in [w*64, w*64+64): i.e. all heads, d-slice [8w, 8w+8)
    {
        v2f a[16];
        #pragma unroll
        for (int ks = 0; ks < 16; ++ks) {
            int col = ks * 4 + 2 * hf;
            v2f t = { s_emb[lm * EMB_RS + col], s_emb[lm * EMB_RS + col + 1] };
            a[ks] = t;
        }
        const int jbase = w * 64;
        const float* Wmat[3] = { Wq, Wk, Wv };
        float*       Smat[3] = { s_q, s_k, s_v };
        #pragma unroll
        for (int m = 0; m < 3; ++m) {
            const float* W = Wmat[m];
            float*       S = Smat[m];
            #pragma unroll
            for (int nt = 0; nt < 4; ++nt) {
                int j = jbase + nt * 16 + lm;        // output column (c*8 + h)
                v8f acc = z8;
                #pragma unroll
                for (int ks = 0; ks < 16; ++ks) {
                    int kr = ks * 4 + 2 * hf;
                    v2f b = { W[(size_t)kr * CH + j], W[(size_t)(kr + 1) * CH + j] };
                    acc = wmma4(a[ks], b, acc);
                }
                int h = j & 7, d = j >> 3;
                #pragma unroll
                for (int r = 0; r < 8; ++r) {
                    int n = r + 8 * hf;              // token row (2 bp stacked)
                    S[h * QKV_HS + n * QKV_RS + d] = acc[r];
                }
            }
        }
    }
    __syncthreads();

    // -------- phase 2: per-head channel attention, per bp --------
    const float* q_h = s_q + w * QKV_HS;
    const float* k_h = s_k + w * QKV_HS;
    const float* v_h = s_v + w * QKV_HS;

    #pragma unroll 1
    for (int bp = 0; bp < 2; ++bp) {
        // T = attn^T : T[d,c] = sum_n k[n,d] * q[n,c]   (A<-k, B<-q, K = n (8))
        v8f T[4][4];
        #pragma unroll
        for (int mt = 0; mt < 4; ++mt)
            #pragma unroll
            for (int nt = 0; nt < 4; ++nt) T[mt][nt] = z8;

        #pragma unroll
        for (int ks = 0; ks < 2; ++ks) {
            int n0 = bp * 8 + ks * 4 + 2 * hf;
            v2f Ak[4], Bq[4];
            #pragma unroll
            for (int mt = 0; mt < 4; ++mt) {
                v2f t = { k_h[n0 * QKV_RS + mt * 16 + lm],
                          k_h[(n0 + 1) * QKV_RS + mt * 16 + lm] };
                Ak[mt] = t;
            }
            #pragma unroll
            for (int nt = 0; nt < 4; ++nt) {
                v2f t = { q_h[n0 * QKV_RS + nt * 16 + lm],
                          q_h[(n0 + 1) * QKV_RS + nt * 16 + lm] };
                Bq[nt] = t;
            }
            #pragma unroll
            for (int mt = 0; mt < 4; ++mt)
                #pragma unroll
                for (int nt = 0; nt < 4; ++nt)
                    T[mt][nt] = wmma4(Ak[mt], Bq[nt], T[mt][nt]);
        }

        // InstanceNorm over the whole 64x64 map (biased var)
        float s = 0.f, ss = 0.f;
        #pragma unroll
        for (int mt = 0; mt < 4; ++mt)
            #pragma unroll
            for (int nt = 0; nt < 4; ++nt)
                #pragma unroll
                for (int r = 0; r < 8; ++r) {
                    float x = T[mt][nt][r];
                    s += x; ss = fmaf(x, x, ss);
                }
        #pragma unroll
        for (int off = 16; off >= 1; off >>= 1) {
            s  += __shfl_xor(s,  off, 32);
            ss += __shfl_xor(ss, off, 32);
        }
        float mu   = s * (1.0f / 4096.0f);
        float var  = fmaf(ss, 1.0f / 4096.0f, -mu * mu);
        float rstd = rsqrtf(var + EPSV);
        #pragma unroll
        for (int mt = 0; mt < 4; ++mt)
            #pragma unroll
            for (int nt = 0; nt < 4; ++nt)
                #pragma unroll
                for (int r = 0; r < 8; ++r)
                    T[mt][nt][r] = (T[mt][nt][r] - mu) * rstd;

        // softmax over d (rows of T): per lane 32 regs + one xor-16 shuffle
        #pragma unroll
        for (int nt = 0; nt < 4; ++nt) {
            float mx = -3.0e38f;
            #pragma unroll
            for (int mt = 0; mt < 4; ++mt)
                #pragma unroll
                for (int r = 0; r < 8; ++r) mx = fmaxf(mx, T[mt][nt][r]);
            mx = fmaxf(mx, __shfl_xor(mx, 16, 32));
            float sum = 0.f;
            #pragma unroll
            for (int mt = 0; mt < 4; ++mt)
                #pragma unroll
                for (int r = 0; r < 8; ++r) {
                    float e = __expf(T[mt][nt][r] - mx);
                    T[mt][nt][r] = e; sum += e;
                }
            sum += __shfl_xor(sum, 16, 32);
            float inv = 1.0f / sum;
            #pragma unroll
            for (int mt = 0; mt < 4; ++mt)
                #pragma unroll
                for (int r = 0; r < 8; ++r) T[mt][nt][r] *= inv;
        }

        // ctx^T : D2[n,c] = sum_d v[n,d] * P^T[d,c]   (B built from T regs, half-swap via shfl)
        v8f D2[4] = { z8, z8, z8, z8 };
        int nrow = bp * 8 + (lm & 7);
        #pragma unroll
        for (int ks = 0; ks < 16; ++ks) {
            int col = ks * 4 + 2 * hf;
            v2f A2 = { v_h[nrow * QKV_RS + col], v_h[nrow * QKV_RS + col + 1] };
            int mt  = ks >> 2;
            int dlo = (ks & 3) * 4;
            #pragma unroll
            for (int nt = 0; nt < 4; ++nt) {
                v2f B2;
                #pragma unroll
                for (int r = 0; r < 2; ++r) {
                    if (dlo < 8) {
                        float t0 = T[mt][nt][dlo + r];
                        float t2 = T[mt][nt][dlo + r + 2];
                        float up = __shfl_xor(t2, 16, 32);
                        B2[r] = hf ? up : t0;
                    } else {
                        float t0 = T[mt][nt][dlo - 8 + r];
                        float t2 = T[mt][nt][dlo - 8 + r + 2];
                        float dn = __shfl_xor(t0, 16, 32);
                        B2[r] = hf ? t2 : dn;
                    }
                }
                D2[nt] = wmma4(A2, B2, D2[nt]);
            }
        }

        // scatter ctx rows into X with (c heads) interleave; valid rows live in lanes 0-15
        if (hf == 0) {
            #pragma unroll
            for (int nt = 0; nt < 4; ++nt)
                #pragma unroll
                for (int r = 0; r < 8; ++r) {
                    int c = nt * 16 + lm;
                    s_x[(bp * 8 + r) * X_RS + c * 8 + w] = D2[nt][r];
                }
        }
    }
    __syncthreads();

    // -------- phase 3: out = X(16x512) @ Wo(512x64), K split across 8 waves --------
    float* s_part = s_pool + OFF_PART;     // [8][16][64], aliases dead emb/q region
    {
        v8f D4[4] = { z8, z8, z8, z8 };
        #pragma unroll
        for (int ks = 0; ks < 16; ++ks) {
            int kk = w * 64 + ks * 4 + 2 * hf;
            v2f A4 = { s_x[lm * X_RS + kk], s_x[lm * X_RS + kk + 1] };
            #pragma unroll
            for (int nt = 0; nt < 4; ++nt) {
                v2f B4 = { Wo[(size_t)kk * CDIM + nt * 16 + lm],
                           Wo[(size_t)(kk + 1) * CDIM + nt * 16 + lm] };
                D4[nt] = wmma4(A4, B4, D4[nt]);
            }
        }
        #pragma unroll
        for (int nt = 0; nt < 4; ++nt)
            #pragma unroll
            for (int r = 0; r < 8; ++r) {
                int row = r + 8 * hf;
                int c   = nt * 16 + lm;
                s_part[w * 1024 + row * 64 + c] = D4[nt][r];
            }
    }
    __syncthreads();

    // -------- phase 4: deterministic K-partial reduction + rearrange scatter --------
    {
        int row = tid >> 4;                  // 0..15 -> (bp_local, n)
        int c0  = (tid & 15) * 4;
        float acc[4] = { 0.f, 0.f, 0.f, 0.f };
        #pragma unroll
        for (int wp = 0; wp < 8; ++wp)
            #pragma unroll
            for (int j = 0; j < 4; ++j)
                acc[j] += s_part[wp * 1024 + row * 64 + c0 + j];

        int bp = blockIdx.x * 2 + (row >> 3);
        int n  = row & 7;
        int b  = bp / 27;
        int g  = bp % 27;
        int dd = g / 9, hh = (g / 3) % 3, ww = g % 3;
        int p1 = n >> 2, p2 = (n >> 1) & 1, p3 = n & 1;
        int t  = dd * 72 + p1 * 36 + hh * 12 + p2 * 6 + ww * 2 + p3;

        float4 v = make_float4(acc[0], acc[1], acc[2], acc[3]);
        *(float4*)(out + ((size_t)b * 216 + t) * CDIM + c0) = v;
    }
}

extern "C" void kernel_launch(void* const* d_in, const int* in_sizes, int n_in,
                              void* d_out, int out_size, void* d_ws, size_t ws_size,
                              hipStream_t stream) {
    (void)in_sizes; (void)n_in; (void)d_ws; (void)ws_size; (void)out_size;
    const float* emb = (const float*)d_in[0];
    const float* Wq  = (const float*)d_in[1];
    const float* Wk  = (const float*)d_in[2];
    const float* Wv  = (const float*)d_in[3];
    const float* Wo  = (const float*)d_in[4];
    float* out = (float*)d_out;
    ca_fused<<<dim3(NBP / 2), dim3(256), 0, stream>>>(emb, Wq, Wk, Wv, Wo, out);
}